// PrimitiveMixerBlock_91216515432857
// MI455X (gfx1250) — compile-verified
//
#include <hip/hip_runtime.h>
#include <hip/hip_bf16.h>
#include <math.h>

// ============================================================================
// ContractiveSequenceMixer for MI455X (gfx1250, wave32, WMMA, TDM)
//   B=32, S=1024, D=1024, fp32 throughout (matches reference precision).
//
// Phase 1 (mixer_gemm_gates_pin): G = sigmoid(x@Wg^T + bg), P = x@Wi^T + bi
//   - 16384 blocks x 256 thr (8 waves), 64x32 output tile per block
//   - double-buffered TDM tensor_load_to_lds staging of X / Wg / Wi chunks
//   - V_WMMA_F32_16X16X4_F32 accumulation (two accumulators share the A tile)
//
// Phase 2 (mixer_scan): persistent 128 single-wave workgroups, 1024 steps
//   - each WG owns a 16(d) x 16(b) tile of the state; Ws slice resident in LDS
//   - per step: TDM-load H tile (= emitted[:, t-1, :] rows) into LDS,
//     K=1024 WMMA GEMM, elementwise gate update, store emitted, grid barrier
//   - TDM pad_enable -> padded LDS row stride to dodge bank conflicts
//
// NOTE: the TDM builtin's LDS write is invisible to alias analysis (the LDS
// address is an integer in the descriptor), so we (a) keep one volatile guard
// store into each shared array so its loads cannot be folded to undef, and
// (b) use asm memory fences around TDM issue/wait to pin load ordering.
//
// Workspace layout: [0,256) barrier counter (memset to 0 each launch),
//   then G (128 MB), then P (128 MB).  Requires ws_size >= 256 + 256 MiB.
// ============================================================================

#define S_LEN 1024
#define D_DIM 1024
#define B_DIM 32

typedef float v2f __attribute__((ext_vector_type(2)));
typedef float v4f __attribute__((ext_vector_type(4)));
typedef float v8f __attribute__((ext_vector_type(8)));
typedef unsigned int u32x4 __attribute__((ext_vector_type(4)));
typedef int i32x4 __attribute__((ext_vector_type(4)));
typedef int i32x8 __attribute__((ext_vector_type(8)));

// Compiler-level memory fence: TDM writes LDS behind the optimizer's back.
#define TDM_MEM_FENCE() asm volatile("" ::: "memory")

// --- TDM descriptor builder: 2D fp32 tile, optional LDS padding -------------
__device__ __forceinline__ void tdm_load_2d_f32(
    unsigned lds_byte_off, const void* gptr,
    unsigned tile0, unsigned tile1,
    unsigned dim0, unsigned dim1,
    unsigned long long stride0_elems,
    unsigned pad_interval_code, unsigned pad_amount_code)
{
    unsigned long long ga = (unsigned long long)(uintptr_t)gptr;
    u32x4 g0;
    g0[0] = 1u;                                              // count=1, user desc
    g0[1] = lds_byte_off;                                    // lds_addr
    g0[2] = (unsigned)ga;                                    // global_addr[31:0]
    g0[3] = ((unsigned)(ga >> 32) & 0x1FFFFFFu) | 0x80000000u; // addr[56:32] | type=2

    unsigned flags = (2u << 16);                             // data_size = 4B
    if (pad_interval_code | pad_amount_code)
        flags |= (1u << 20) | (pad_interval_code << 22) | (pad_amount_code << 25);
    i32x8 g1;
    g1[0] = (int)flags;
    g1[1] = (int)((dim0 & 0xFFFFu) << 16);                   // tensor_dim0 lo16
    g1[2] = (int)((dim0 >> 16) | ((dim1 & 0xFFFFu) << 16));  // dim0 hi16 | dim1 lo16
    g1[3] = (int)((dim1 >> 16) | ((tile0 & 0xFFFFu) << 16)); // dim1 hi16 | tile_dim0
    g1[4] = (int)(tile1 & 0xFFFFu);                          // tile_dim1 | tile_dim2=0
    g1[5] = (int)(unsigned)(stride0_elems & 0xFFFFFFFFull);  // tensor_dim0_stride lo32
    g1[6] = (int)((unsigned)(stride0_elems >> 32) & 0xFFFFu);// stride0 hi16 | stride1 lo=0
    g1[7] = 0;
    i32x4 z4 = (i32x4){0, 0, 0, 0};
    i32x8 z8 = (i32x8){0, 0, 0, 0, 0, 0, 0, 0};
    TDM_MEM_FENCE();   // keep prior LDS reads of the target buffer above the issue
    // 6-arg form (clang-23 / therock-10.0 headers): groups 0..3 + extra group + cpol
    __builtin_amdgcn_tensor_load_to_lds(g0, g1, z4, z4, z8, 0);
}

__device__ __forceinline__ void tdm_wait_all() {
    __builtin_amdgcn_s_wait_tensorcnt(0);
    TDM_MEM_FENCE();   // LDS reads below must not hoist above this point
}

// ============================================================================
// Phase 1: G = sigmoid(x @ Wg^T + bg), P = x @ Wi^T + bi
// ============================================================================
#define XS_DW  (64 * 68)                 // 64x64 X chunk, padded rows (stride 68)
#define WT_DW  (32 * 68)                 // 32x64 W chunk, padded rows
#define BUF_DW (XS_DW + 2 * WT_DW)       // 8704 dwords per buffer

__global__ __launch_bounds__(256) void mixer_gemm_gates_pin(
    const float* __restrict__ x,  const float* __restrict__ Wg,
    const float* __restrict__ Wi, const float* __restrict__ bg,
    const float* __restrict__ bi, float* __restrict__ G, float* __restrict__ P)
{
    __shared__ float sm[2 * BUF_DW + 4]; // +4: guard pad, base LDS offset 0

    // Guard store: gives the LDS global a real writer so ds loads survive.
    if (threadIdx.x == 0) *(volatile float*)&sm[2 * BUF_DW] = 0.0f;

    const int lane = threadIdx.x & 31;
    const int w    = threadIdx.x >> 5;
    const int half = lane >> 4;          // 0/1 : K sub-pair per WMMA layout
    const int lm   = lane & 15;
    const int mw   = w & 3;              // 4 M sub-tiles
    const int nw   = w >> 2;             // 2 N sub-tiles
    const size_t m0 = (size_t)blockIdx.x * 64;   // row tile over flattened (b,s)
    const size_t n0 = (size_t)blockIdx.y * 32;   // col tile over output dim e

    v8f accG = {};
    v8f accP = {};

    if (w == 0) {   // prime chunk 0 (TDM ignores EXEC; one issue per wave)
        tdm_load_2d_f32(0,                 x  + m0 * D_DIM, 64, 64, D_DIM, 64, D_DIM, 5, 3);
        tdm_load_2d_f32(XS_DW * 4,         Wg + n0 * D_DIM, 64, 32, D_DIM, 32, D_DIM, 5, 3);
        tdm_load_2d_f32((XS_DW + WT_DW)*4, Wi + n0 * D_DIM, 64, 32, D_DIM, 32, D_DIM, 5, 3);
    }

    for (int c = 0; c < 16; ++c) {
        if (w == 0) tdm_wait_all();
        __syncthreads();
        if (w == 0 && c + 1 < 16) {      // async prefetch next chunk into other buffer
            unsigned boff = (unsigned)(((c + 1) & 1) * BUF_DW * 4);
            int k0 = (c + 1) * 64;
            tdm_load_2d_f32(boff,                   x  + m0 * D_DIM + k0, 64, 64, D_DIM, 64, D_DIM, 5, 3);
            tdm_load_2d_f32(boff + XS_DW * 4,       Wg + n0 * D_DIM + k0, 64, 32, D_DIM, 32, D_DIM, 5, 3);
            tdm_load_2d_f32(boff + (XS_DW+WT_DW)*4, Wi + n0 * D_DIM + k0, 64, 32, D_DIM, 32, D_DIM, 5, 3);
        }
        const float* xs  = &sm[(c & 1) * BUF_DW];
        const float* wgs = xs + XS_DW;
        const float* wis = wgs + WT_DW;
        const int ar = (mw * 16 + lm) * 68;   // A rows (x)
        const int br = (nw * 16 + lm) * 68;   // B rows (W, accessed as B[k][n]=W[n][k])
#pragma unroll
        for (int kk = 0; kk < 64; kk += 4) {
            const int k2 = kk + 2 * half;
            v2f a  = *(const v2f*)(xs  + ar + k2);
            v2f bG = *(const v2f*)(wgs + br + k2);
            v2f bI = *(const v2f*)(wis + br + k2);
            accG = __builtin_amdgcn_wmma_f32_16x16x4_f32(false, a, false, bG, (short)0, accG, false, false);
            accP = __builtin_amdgcn_wmma_f32_16x16x4_f32(false, a, false, bI, (short)0, accP, false, false);
        }
        __syncthreads();
    }

    const size_t n = n0 + (size_t)nw * 16 + lm;
    const float bgn = bg[n];
    const float bin = bi[n];
#pragma unroll
    for (int v = 0; v < 8; ++v) {        // C layout: lane L,vgpr v -> m = v + 8*(L/16)
        const size_t m = m0 + (size_t)mw * 16 + v + 8 * half;
        const float gpre = accG[v] + bgn;
        G[m * D_DIM + n] = 1.0f / (1.0f + __expf(-gpre));
        P[m * D_DIM + n] = accP[v] + bin;
    }
}

// ============================================================================
// Phase 2: persistent scan. 128 blocks x 32 threads (1 wave each).
// ============================================================================
#define ROW_DW 1032                       // 1024 + padding (2 dw per 256 dw)
__device__ __forceinline__ int pcol(int c) { return c + ((c >> 8) << 1); }

__global__ __launch_bounds__(32) void mixer_scan(
    const float* __restrict__ G, const float* __restrict__ P,
    const float* __restrict__ Ws, const float* __restrict__ bs,
    float* __restrict__ out, unsigned* __restrict__ cnt)
{
    __shared__ float sm[2 * 16 * ROW_DW + 4]; // Ws slice @0, H tile @16*ROW_DW, +guard

    // Guard store: gives the LDS global a real writer so ds loads survive.
    if (threadIdx.x == 0) *(volatile float*)&sm[2 * 16 * ROW_DW] = 0.0f;

    const int lane = threadIdx.x;
    const int half = lane >> 4;
    const int lm   = lane & 15;
    const int d0   = (blockIdx.x >> 1) * 16;   // 64 d-tiles
    const int b0   = (blockIdx.x & 1) * 16;    // 2  b-tiles
    const unsigned nblk = gridDim.x;

    // Resident Ws slice: rows d0..d0+15, K = 0..1023 (64KB + pad)
    tdm_load_2d_f32(0, Ws + (size_t)d0 * D_DIM, 1024, 16, D_DIM, 16, D_DIM, 7, 1);

    const int dl = d0 + 8 * half;              // this lane's 8 consecutive d's
    const v4f bs_lo = *(const v4f*)(bs + dl);
    const v4f bs_hi = *(const v4f*)(bs + dl + 4);

    const float* wsr = sm;                     // A rows: Ws[d][k]
    const float* hsr = sm + 16 * ROW_DW;       // B rows: H^T[b][k]

    v8f h = {};                                // our (d,b) state tile, h_0 = 0
    tdm_wait_all();

    for (int t = 0; t < S_LEN; ++t) {
        v8f mix = {};
        if (t > 0) {
            // H tile = emitted rows b0..b0+15 at step t-1 (row stride S*D elems)
            tdm_load_2d_f32((unsigned)(16 * ROW_DW * 4),
                            out + ((size_t)b0 * S_LEN + (size_t)(t - 1)) * D_DIM,
                            1024, 16, D_DIM, 16,
                            (unsigned long long)S_LEN * D_DIM, 7, 1);
            tdm_wait_all();
            v8f a0 = {}, a1 = {};
            const int arow = lm * ROW_DW;
#pragma unroll 4
            for (int kk = 0; kk < D_DIM; kk += 8) {
                const int ka = pcol(kk + 2 * half);
                const int kb = pcol(kk + 4 + 2 * half);
                v2f aA = *(const v2f*)(wsr + arow + ka);
                v2f bA = *(const v2f*)(hsr + arow + ka);
                a0 = __builtin_amdgcn_wmma_f32_16x16x4_f32(false, aA, false, bA, (short)0, a0, false, false);
                v2f aB = *(const v2f*)(wsr + arow + kb);
                v2f bB = *(const v2f*)(hsr + arow + kb);
                a1 = __builtin_amdgcn_wmma_f32_16x16x4_f32(false, aB, false, bB, (short)0, a1, false, false);
            }
            mix = a0 + a1;
        }

        // gate/pin for this (d,b,t) tile: 8 consecutive d's per lane
        const size_t base = ((size_t)(b0 + lm) * S_LEN + (size_t)t) * D_DIM + dl;
        v4f g_lo = *(const v4f*)(G + base);
        v4f g_hi = *(const v4f*)(G + base + 4);
        v4f p_lo = *(const v4f*)(P + base);
        v4f p_hi = *(const v4f*)(P + base + 4);

        v4f h_lo, h_hi;
#pragma unroll
        for (int v = 0; v < 4; ++v) {
            float m_lo = mix[v]     + bs_lo[v] + p_lo[v];
            float m_hi = mix[v + 4] + bs_hi[v] + p_hi[v];
            float n_lo = g_lo[v] * m_lo + (1.0f - g_lo[v]) * h[v];
            float n_hi = g_hi[v] * m_hi + (1.0f - g_hi[v]) * h[v + 4];
            h[v] = n_lo;  h[v + 4] = n_hi;
            h_lo[v] = n_lo; h_hi[v] = n_hi;
        }

        *(v4f*)(out + base)     = h_lo;        // emitted[b, t, d]
        *(v4f*)(out + base + 4) = h_hi;
        if (t == S_LEN - 1) {                  // final_state appended after emitted
            float* fs = out + (size_t)B_DIM * S_LEN * D_DIM + (size_t)(b0 + lm) * D_DIM + dl;
            *(v4f*)fs       = h_lo;
            *(v4f*)(fs + 4) = h_hi;
        }

        // grid-wide barrier: all tiles of step t visible before step t+1 TDM reads
        __threadfence();
        if (lane == 0) {
            __hip_atomic_fetch_add(cnt, 1u, __ATOMIC_RELEASE, __HIP_MEMORY_SCOPE_AGENT);
            const unsigned tgt = (unsigned)(t + 1) * nblk;
            while (__hip_atomic_load(cnt, __ATOMIC_ACQUIRE, __HIP_MEMORY_SCOPE_AGENT) < tgt)
                __builtin_amdgcn_s_sleep(1);
        }
        __threadfence();
    }
}

// ============================================================================
extern "C" void kernel_launch(void* const* d_in, const int* in_sizes, int n_in,
                              void* d_out, int out_size, void* d_ws, size_t ws_size,
                              hipStream_t stream) {
    const float* x  = (const float*)d_in[0];
    const float* Wg = (const float*)d_in[1];
    const float* bg = (const float*)d_in[2];
    const float* Ws = (const float*)d_in[3];
    const float* bs = (const float*)d_in[4];
    const float* Wi = (const float*)d_in[5];
    const float* bi = (const float*)d_in[6];
    float* out = (float*)d_out;

    char* ws = (char*)d_ws;
    unsigned* cnt = (unsigned*)ws;                       // barrier counter
    float* G = (float*)(ws + 256);                       // 32768 x 1024 fp32
    float* P = G + (size_t)B_DIM * S_LEN * D_DIM;        // 32768 x 1024 fp32

    // Deterministic: barrier counter reset each launch (graph-capture legal).
    (void)hipMemsetAsync(ws, 0, 256, stream);

    dim3 g1(B_DIM * S_LEN / 64, D_DIM / 32, 1);          // 512 x 32 blocks
    mixer_gemm_gates_pin<<<g1, 256, 0, stream>>>(x, Wg, Wi, bg, bi, G, P);

    mixer_scan<<<128, 32, 0, stream>>>(G, P, Ws, bs, out, cnt);
}